// GATZinc_283467842550
// MI455X (gfx1250) — compile-verified
//
#include <hip/hip_runtime.h>
#include <hip/hip_bf16.h>

#define NN 10000
#define DD 256
#define EE 320000
#define ET 330000   // edges + self loops
#define LDSW 260    // padded LDS row stride (floats): rows land on distinct banks

typedef float v2f __attribute__((ext_vector_type(2)));
typedef float v8f __attribute__((ext_vector_type(8)));
typedef int   v4i __attribute__((ext_vector_type(4)));

#if defined(__has_builtin)
#if __has_builtin(__builtin_amdgcn_global_load_async_to_lds_b128) && \
    __has_builtin(__builtin_amdgcn_s_wait_asynccnt)
#define USE_ASYNC_LDS 1
#endif
#endif
#ifndef USE_ASYNC_LDS
#define USE_ASYNC_LDS 0
#endif

// ---- ordered-uint mapping for float atomicMax ----
__device__ __forceinline__ unsigned f2o(float f) {
    unsigned b = __float_as_uint(f);
    return (b & 0x80000000u) ? ~b : (b | 0x80000000u);
}
__device__ __forceinline__ float o2f(unsigned u) {
    return (u & 0x80000000u) ? __uint_as_float(u & 0x7fffffffu)
                             : __uint_as_float(~u);
}
#define ORD_NEG_INF 0x007FFFFFu   // f2o(-inf)

// ---- h[n,:] = emb[x[n],:] ----
__global__ void gat_embed(const int* __restrict__ x, const float* __restrict__ emb,
                          float* __restrict__ h) {
    int t = blockIdx.x * blockDim.x + threadIdx.x;
    if (t < NN * DD) {
        int n = t >> 8, d = t & 255;
        h[t] = emb[x[n] * DD + d];
    }
}

// ---- C = A(10000x256) @ B(256x256) via V_WMMA_F32_16X16X4_F32 ----
// grid 625, block 128 (4 waves). Block: 16-row strip x all 256 cols.
// A tile (16x256) staged in LDS (async copy); each wave owns 4 N-tiles and
// reuses one A fragment across 4 WMMAs per k-step.
__global__ void __launch_bounds__(128) gat_gemm_wmma(const float* __restrict__ A,
                                                     const float* __restrict__ B,
                                                     float* __restrict__ C) {
    __shared__ __align__(16) float As[16 * LDSW];

    const int t    = threadIdx.x;
    const int m0   = blockIdx.x * 16;
    const int lane = t & 31;
    const int wave = t >> 5;
    const int half = lane >> 4;   // 0/1
    const int lm   = lane & 15;

    // ---- stage A tile: 16 rows x 256 cols = 1024 float4 chunks, 8 per thread
    const float* gbase = A + (size_t)m0 * DD;
#pragma unroll
    for (int i = 0; i < 8; ++i) {
        int c   = i * 128 + t;          // chunk id
        int row = c >> 6;
        int col = (c & 63) << 2;
        float*       lp = &As[row * LDSW + col];
        const float* gp = gbase + row * DD + col;
#if USE_ASYNC_LDS
        __builtin_amdgcn_global_load_async_to_lds_b128(
            (v4i*)gp,
            (__attribute__((address_space(3))) v4i*)lp, 0, 0);
#else
        *(float4*)lp = *(const float4*)gp;
#endif
    }
#if USE_ASYNC_LDS
    __builtin_amdgcn_s_wait_asynccnt(0);
#endif
    __syncthreads();

    // ---- 4 accumulators per wave: cols wave*64 + {0,16,32,48}
    const int  cbase0 = wave * 64 + lm;
    const float* arow = &As[lm * LDSW];
    v8f acc0 = {}, acc1 = {}, acc2 = {}, acc3 = {};

#pragma unroll 2
    for (int k0 = 0; k0 < DD; k0 += 4) {
        const int ka = k0 + half * 2;
        v2f a = *(const v2f*)(arow + ka);               // A[m][ka..ka+1] from LDS
        const float* br0 = B + (size_t)ka * DD + cbase0;
        const float* br1 = br0 + DD;
        v2f b0, b1, b2, b3;
        b0.x = br0[0];  b0.y = br1[0];
        b1.x = br0[16]; b1.y = br1[16];
        b2.x = br0[32]; b2.y = br1[32];
        b3.x = br0[48]; b3.y = br1[48];
        acc0 = __builtin_amdgcn_wmma_f32_16x16x4_f32(false, a, false, b0, (short)0, acc0, false, false);
        acc1 = __builtin_amdgcn_wmma_f32_16x16x4_f32(false, a, false, b1, (short)0, acc1, false, false);
        acc2 = __builtin_amdgcn_wmma_f32_16x16x4_f32(false, a, false, b2, (short)0, acc2, false, false);
        acc3 = __builtin_amdgcn_wmma_f32_16x16x4_f32(false, a, false, b3, (short)0, acc3, false, false);
    }

    float* cb = C + (size_t)(m0 + half * 8) * DD + cbase0;
#pragma unroll
    for (int r = 0; r < 8; ++r) {
        cb[(size_t)r * DD +  0] = acc0[r];
        cb[(size_t)r * DD + 16] = acc1[r];
        cb[(size_t)r * DD + 32] = acc2[r];
        cb[(size_t)r * DD + 48] = acc3[r];
    }
}

// ---- alpha_s[n] = hW[n,:]·a_src, alpha_d[n] = hW[n,:]·a_dst ----
__global__ void gat_alpha_dots(const float* __restrict__ hw,
                               const float* __restrict__ a_src,
                               const float* __restrict__ a_dst,
                               float* __restrict__ as, float* __restrict__ ad) {
    __shared__ float ss[DD], sd[DD];
    int n = blockIdx.x, t = threadIdx.x;
    float hv = hw[(size_t)n * DD + t];
    ss[t] = hv * a_src[t];
    sd[t] = hv * a_dst[t];
    __syncthreads();
    for (int s = DD / 2; s > 0; s >>= 1) {
        if (t < s) { ss[t] += ss[t + s]; sd[t] += sd[t + s]; }
        __syncthreads();
    }
    if (t == 0) { as[n] = ss[0]; ad[n] = sd[0]; }
}

// ---- per-layer init: accum = bias broadcast, max = -inf, denom = 0 ----
__global__ void gat_init(float* __restrict__ accum, const float* __restrict__ bias,
                         unsigned* __restrict__ maxo, float* __restrict__ denom) {
    int t = blockIdx.x * blockDim.x + threadIdx.x;
    if (t < NN * DD) accum[t] = bias[t & 255];
    if (t < NN) { maxo[t] = ORD_NEG_INF; denom[t] = 0.0f; }
}

__device__ __forceinline__ void edge_sd(const int* __restrict__ ei, int e, int& s, int& d) {
    if (e < EE) { s = ei[e]; d = ei[EE + e]; }
    else        { s = d = e - EE; }          // self loops appended
}

// ---- pass 1: logits + segment max ----
__global__ void gat_edge_logit(const int* __restrict__ ei,
                               const float* __restrict__ as, const float* __restrict__ ad,
                               float* __restrict__ elog, unsigned* __restrict__ maxo) {
    int e = blockIdx.x * blockDim.x + threadIdx.x;
    if (e >= ET) return;
    int s, d; edge_sd(ei, e, s, d);
    float l = as[s] + ad[d];
    l = (l > 0.0f) ? l : 0.2f * l;           // leaky_relu
    elog[e] = l;
    atomicMax(&maxo[d], f2o(l));
}

// ---- pass 2: exp + segment sum ----
__global__ void gat_edge_exp(const int* __restrict__ ei,
                             float* __restrict__ elog, const unsigned* __restrict__ maxo,
                             float* __restrict__ denom) {
    int e = blockIdx.x * blockDim.x + threadIdx.x;
    if (e >= ET) return;
    int s, d; edge_sd(ei, e, s, d);
    float ex = __expf(elog[e] - o2f(maxo[d]));
    elog[e] = ex;
    atomicAdd(&denom[d], ex);
}

// ---- pass 2b: alpha = e / denom[dst] (once per edge) ----
__global__ void gat_edge_norm(const int* __restrict__ ei,
                              float* __restrict__ elog, const float* __restrict__ denom) {
    int e = blockIdx.x * blockDim.x + threadIdx.x;
    if (e >= ET) return;
    int s, d; edge_sd(ei, e, s, d);
    elog[e] = elog[e] / denom[d];
}

// ---- pass 3: accum[dst,:] += alpha * hW[src,:]  (64 threads/edge, float4) ----
__global__ void gat_edge_aggregate(const int* __restrict__ ei,
                                   const float* __restrict__ elog,
                                   const float* __restrict__ hw,
                                   float* __restrict__ accum) {
    int t = blockIdx.x * blockDim.x + threadIdx.x;
    int e = t >> 6;
    if (e >= ET) return;
    int q = (t & 63) << 2;
    int s, d; edge_sd(ei, e, s, d);
    float alpha = elog[e];
    float4 hv = *(const float4*)(hw + (size_t)s * DD + q);
    float* o = accum + (size_t)d * DD + q;
    atomicAdd(o + 0, alpha * hv.x);
    atomicAdd(o + 1, alpha * hv.y);
    atomicAdd(o + 2, alpha * hv.z);
    atomicAdd(o + 3, alpha * hv.w);
}

// ---- h = relu(accum) + h ----
__global__ void gat_relu_residual(const float* __restrict__ accum, float* __restrict__ h) {
    int t = blockIdx.x * blockDim.x + threadIdx.x;
    if (t < NN * DD) {
        float v = accum[t];
        h[t] = (v > 0.0f ? v : 0.0f) + h[t];
    }
}

extern "C" void kernel_launch(void* const* d_in, const int* in_sizes, int n_in,
                              void* d_out, int out_size, void* d_ws, size_t ws_size,
                              hipStream_t stream) {
    (void)in_sizes; (void)n_in; (void)out_size; (void)ws_size;

    const int*   x   = (const int*)d_in[0];
    const int*   ei  = (const int*)d_in[1];   // [2, EE]
    const float* emb = (const float*)d_in[3];
    const float* W[3]   = {(const float*)d_in[4],  (const float*)d_in[8],  (const float*)d_in[12]};
    const float* asr[3] = {(const float*)d_in[5],  (const float*)d_in[9],  (const float*)d_in[13]};
    const float* adt[3] = {(const float*)d_in[6],  (const float*)d_in[10], (const float*)d_in[14]};
    const float* bia[3] = {(const float*)d_in[7],  (const float*)d_in[11], (const float*)d_in[15]};

    float* h = (float*)d_out;                 // persistent node features, final output

    // workspace layout (floats)
    float*    hw     = (float*)d_ws;             // [NN*DD]
    float*    accum  = hw + (size_t)NN * DD;     // [NN*DD]
    float*    alphas = accum + (size_t)NN * DD;  // [NN]
    float*    alphad = alphas + NN;              // [NN]
    unsigned* maxo   = (unsigned*)(alphad + NN); // [NN]
    float*    denom  = (float*)(maxo + NN);      // [NN]
    float*    elog   = denom + NN;               // [ET]

    const int TPB  = 256;
    const int gND  = (NN * DD + TPB - 1) / TPB;
    const int gE   = (ET + TPB - 1) / TPB;
    const int gE64 = (ET * 64 + TPB - 1) / TPB;

    gat_embed<<<gND, TPB, 0, stream>>>(x, emb, h);

    for (int l = 0; l < 3; ++l) {
        gat_gemm_wmma<<<NN / 16, 128, 0, stream>>>(h, W[l], hw);
        gat_alpha_dots<<<NN, DD, 0, stream>>>(hw, asr[l], adt[l], alphas, alphad);
        gat_init<<<gND, TPB, 0, stream>>>(accum, bia[l], maxo, denom);
        gat_edge_logit<<<gE, TPB, 0, stream>>>(ei, alphas, alphad, elog, maxo);
        gat_edge_exp<<<gE, TPB, 0, stream>>>(ei, elog, maxo, denom);
        gat_edge_norm<<<gE, TPB, 0, stream>>>(ei, elog, denom);
        gat_edge_aggregate<<<gE64, TPB, 0, stream>>>(ei, elog, hw, accum);
        gat_relu_residual<<<gND, TPB, 0, stream>>>(accum, h);
    }
}